// fism_79525614452998
// MI455X (gfx1250) — compile-verified
//
#include <hip/hip_runtime.h>
#include <math.h>

#define F_ALPHA 0.5f
#define F_BATA  0.01f
#define F_LAMDA 0.01f
#define F_GAMA  0.01f
#define N_B     4096
#define N_NEG   50
#define DIMK    64
#define TOTAL_H 204800

typedef __attribute__((ext_vector_type(2))) float v2f;
typedef __attribute__((ext_vector_type(8))) float v8f;

__device__ __forceinline__ float sigmoidf(float x) {
    return 1.0f / (1.0f + expf(-x));
}

// ---------------------------------------------------------------------------
// Kernel 1: ragged segment-sum  user_embeds[b][d] = sum_{i in seg b} pu[hist_items[i]][d]
// One 64-thread block per user; binary search the sorted segids for the range.
// ---------------------------------------------------------------------------
__global__ __launch_bounds__(64)
void fism_user_embeds(const float* __restrict__ pu,
                      const int*   __restrict__ hist_items,
                      const int*   __restrict__ hist_segids,
                      float*       __restrict__ ue)
{
    const int b = blockIdx.x;
    const int d = threadIdx.x;                 // 0..63

    // lower_bound(segids, b) and lower_bound(segids, b+1) — uniform across block
    int l = 0, r = TOTAL_H;
    while (l < r) { int m = (l + r) >> 1; if (hist_segids[m] <  b)     l = m + 1; else r = m; }
    const int lo = l;
    r = TOTAL_H;
    while (l < r) { int m = (l + r) >> 1; if (hist_segids[m] <  b + 1) l = m + 1; else r = m; }
    const int hi = l;

    float acc = 0.0f;
    for (int i = lo; i < hi; ++i) {
        const int it = hist_items[i];
        if (i + 1 < hi)
            __builtin_prefetch(&pu[(size_t)hist_items[i + 1] * DIMK + d], 0, 0);
        acc += pu[(size_t)it * DIMK + d];
    }
    ue[(size_t)b * DIMK + d] = acc;
}

// ---------------------------------------------------------------------------
// Kernel 2: per-user scoring using V_WMMA_F32_16X16X4_F32.
// One wave32 per user, 4 waves per block.  Slots 0..49 = negatives,
// slot 50 = positive, 51..63 = padding (ignored).
// ---------------------------------------------------------------------------
__global__ __launch_bounds__(128)
void fism_score(const float* __restrict__ bu,
                const float* __restrict__ bi,
                const float* __restrict__ qi,
                const float* __restrict__ ue,
                const float* __restrict__ uin,
                const int*   __restrict__ users,
                const int*   __restrict__ pos_items,
                const int*   __restrict__ neg_items,
                float*       __restrict__ partial)
{
    __shared__ float u_lds[4][DIMK];
    __shared__ float red[4];

    const int wave = threadIdx.x >> 5;
    const int lane = threadIdx.x & 31;
    const int b    = blockIdx.x * 4 + wave;

    // stage user vector in LDS; each lane covers dims lane and lane+32
    const float x0 = ue[(size_t)b * DIMK + lane];
    const float x1 = ue[(size_t)b * DIMK + lane + 32];
    u_lds[wave][lane]      = x0;
    u_lds[wave][lane + 32] = x1;
    __syncthreads();

    const float t    = 1.0f / sqrtf(uin[b]);          // n^{-ALPHA}, ALPHA = 0.5
    const float buv  = bu[users[b]];
    const int   l16  = lane & 15;
    const int   hi16 = lane >> 4;                     // 0: K pair {0,1}, 1: K pair {2,3}
    const float* uw  = u_lds[wave];

    // regularizer: BATA * ||u||^2 (each lane holds 2 distinct dims)
    float contrib = F_BATA * (x0 * x0 + x1 * x1);

    for (int c = 0; c < 4; ++c) {
        const int s = c * 16 + l16;                   // slot this lane's B column holds
        int it;
        if      (s < N_NEG)  it = neg_items[b * N_NEG + s];
        else if (s == N_NEG) it = pos_items[b];
        else                 it = 0;                  // padded slot, valid row, ignored
        const float* qrow = qi + (size_t)it * DIMK;

        v8f acc = {0.f, 0.f, 0.f, 0.f, 0.f, 0.f, 0.f, 0.f};
        #pragma unroll
        for (int kk = 0; kk < 16; ++kk) {
            const int kb = kk * 4 + hi16 * 2;
            v2f a;  a.x  = uw[kb];   a.y  = uw[kb + 1];     // A: user vec, replicated rows
            const float2 q2 = *(const float2*)(qrow + kb);  // B: 16 item rows, striped
            v2f bm; bm.x = q2.x; bm.y = q2.y;
            acc = __builtin_amdgcn_wmma_f32_16x16x4_f32(
                      false, a, false, bm, (short)0, acc, false, false);
        }

        // D[0][n] (lanes 0..15 of acc[0]) = dot(u, q_slot); rows identical, so
        // every lane's acc[0] is the dot for slot c*16 + (lane&15).
        const float dotv = acc[0];
        if (hi16 == 0) {
            if (s < N_NEG) {
                const float sc = sigmoidf(t * dotv + bi[it] + buv);
                contrib += sc * sc;
            } else if (s == N_NEG) {
                const float sc = sigmoidf(t * dotv + bi[it] + buv);
                const float e  = 1.0f - sc;
                contrib += e * e;
                const float biv = bi[it];
                contrib += F_LAMDA * (biv * biv + F_GAMA * buv * buv);
                float qn = 0.f;                        // BATA * ||q_pos||^2
                for (int d0 = 0; d0 < DIMK; ++d0) { const float q = qrow[d0]; qn += q * q; }
                contrib += F_BATA * qn;
            }
        }
    }

    // wave32 reduction
    #pragma unroll
    for (int off = 16; off > 0; off >>= 1)
        contrib += __shfl_down(contrib, off, 32);
    if (lane == 0) red[wave] = contrib;
    __syncthreads();
    if (threadIdx.x == 0)
        partial[blockIdx.x] = red[0] + red[1] + red[2] + red[3];
}

// ---------------------------------------------------------------------------
// Kernel 3: deterministic fixed-order reduction of 1024 partials -> scalar.
// ---------------------------------------------------------------------------
__global__ __launch_bounds__(256)
void fism_reduce(const float* __restrict__ partial, float* __restrict__ out)
{
    __shared__ float sh[256];
    float s = 0.f;
    for (int i = threadIdx.x; i < N_B / 4; i += 256) s += partial[i];
    sh[threadIdx.x] = s;
    __syncthreads();
    for (int off = 128; off > 0; off >>= 1) {
        if (threadIdx.x < off) sh[threadIdx.x] += sh[threadIdx.x + off];
        __syncthreads();
    }
    if (threadIdx.x == 0) out[0] = sh[0];
}

// ---------------------------------------------------------------------------
extern "C" void kernel_launch(void* const* d_in, const int* in_sizes, int n_in,
                              void* d_out, int out_size, void* d_ws, size_t ws_size,
                              hipStream_t stream)
{
    const float* bu         = (const float*)d_in[0];
    const float* bi         = (const float*)d_in[1];
    const float* qi         = (const float*)d_in[2];
    const float* pu         = (const float*)d_in[3];
    const float* uin        = (const float*)d_in[4];
    const int*   users      = (const int*)  d_in[5];
    const int*   pos_items  = (const int*)  d_in[6];
    const int*   neg_items  = (const int*)  d_in[7];
    const int*   hist_items = (const int*)  d_in[8];
    const int*   hist_seg   = (const int*)  d_in[9];

    float* ue      = (float*)d_ws;                                     // 4096*64 floats = 1 MB
    float* partial = (float*)((char*)d_ws + (size_t)N_B * DIMK * sizeof(float)); // 1024 floats

    fism_user_embeds<<<N_B, 64, 0, stream>>>(pu, hist_items, hist_seg, ue);
    fism_score<<<N_B / 4, 128, 0, stream>>>(bu, bi, qi, ue, uin,
                                            users, pos_items, neg_items, partial);
    fism_reduce<<<1, 256, 0, stream>>>(partial, (float*)d_out);
}